// CompGCNCov_14370960573155
// MI455X (gfx1250) — compile-verified
//
#include <hip/hip_runtime.h>

// CompGCN layer for gfx1250 (MI455X).
// ccorr(a,b) = a @ C(b) with C(b)[j,k]=b[(j+k)%128], so per-edge work becomes
// x[src] @ M[dir,type] with M = C(rel) @ W precomputed (200 matrices, 13MB,
// L2-resident). Edges are bucket-sorted by (dir,type) and processed 32 at a
// time with V_WMMA_F32_16X16X4_F32 (exact fp32 -> matches the fp32 reference;
// the kernel is L2-gather/atomic bound, so low-precision WMMA formats gain
// nothing). M is stored directly in WMMA B-fragment order so each lane's B
// fragment is one coalesced global_load_b64 with immediate offsets.
// Aggregation uses hardware global_atomic_add_f32 into d_out; BatchNorm is
// applied in place (bias and the /3 fold into BN scale/shift).

typedef __attribute__((ext_vector_type(2))) float v2f;
typedef __attribute__((ext_vector_type(8))) float v8f;

#define D 128

struct Tile { int key; int start; int cnt; int pad; };

// B-fragment layout for V_WMMA_F32_16X16X4_F32:
// chunk (kb,nb) holds the 4x16 block K=[4kb,4kb+4) x N=[16nb,16nb+16),
// 64 floats stored contiguously as lane-major {v0,v1} pairs:
//   lane = ((k&3)>>1)*16 + (n&15),  v = k&1
__device__ __forceinline__ size_t bfragIdx(int k, int n) {
  return (size_t)(((k >> 2) * 8 + (n >> 4)) * 64 +
                  (((k >> 1) & 1) * 16 + (n & 15)) * 2 + (k & 1));
}

// ---------------------------------------------------------------- utilities
__global__ void zero_f32(float* __restrict__ p, long n) {
  long i = (long)blockIdx.x * blockDim.x + threadIdx.x;
  long s = (long)gridDim.x * blockDim.x;
  for (; i < n; i += s) p[i] = 0.f;
}

// ------------------------------------------------- bucket sort by (dir,type)
__global__ void hist_kernel(const int* __restrict__ etype, int E, int half,
                            int nrel, int* __restrict__ cnt) {
  int i = blockIdx.x * blockDim.x + threadIdx.x;
  if (i < E) {
    int key = etype[i] + (i >= half ? nrel : 0);
    atomicAdd(&cnt[key], 1);
  }
}

__global__ void scan_kernel(const int* __restrict__ cnt, int nkey,
                            int* __restrict__ offsets, int* __restrict__ tbase,
                            int* __restrict__ numTiles) {
  if (threadIdx.x == 0) {
    int off = 0, toff = 0;
    for (int b = 0; b < nkey; ++b) {
      offsets[b] = off;
      tbase[b] = toff;
      off += cnt[b];
      toff += (cnt[b] + 31) >> 5;
    }
    offsets[nkey] = off;
    *numTiles = toff;
  }
}

__global__ void tiles_kernel(const int* __restrict__ cnt,
                             const int* __restrict__ offsets,
                             const int* __restrict__ tbase,
                             Tile* __restrict__ tiles) {
  int b = blockIdx.x;
  int c = cnt[b];
  int nt = (c + 31) >> 5;
  for (int ti = threadIdx.x; ti < nt; ti += blockDim.x) {
    Tile t;
    t.key = b;
    t.start = offsets[b] + ti * 32;
    int rem = c - ti * 32;
    t.cnt = rem < 32 ? rem : 32;
    t.pad = 0;
    tiles[tbase[b] + ti] = t;
  }
}

__global__ void scatter_kernel(const int* __restrict__ etype, int E, int half,
                               int nrel, const int* __restrict__ offsets,
                               int* __restrict__ cursor, int* __restrict__ sorted) {
  int i = blockIdx.x * blockDim.x + threadIdx.x;
  if (i < E) {
    int key = etype[i] + (i >= half ? nrel : 0);
    int pos = atomicAdd(&cursor[key], 1);
    sorted[offsets[key] + pos] = i;
  }
}

// ------------- M[key] = C(rel) @ W  (+ Mloop), stored in B-fragment order
__global__ __launch_bounds__(128) void buildM_kernel(
    const float* __restrict__ rel, const float* __restrict__ in_w,
    const float* __restrict__ out_w, const float* __restrict__ loop_rel,
    const float* __restrict__ loop_w, int nrel,
    float* __restrict__ Mmat, float* __restrict__ Mloop) {
  int mat = blockIdx.x >> 7;   // which matrix
  int j = blockIdx.x & 127;    // M row index (the K dimension of the GEMM)
  const float* r;
  const float* W;
  float* out;
  if (mat < nrel)            { r = rel + (size_t)mat * D;          W = in_w;   out = Mmat + (size_t)mat * D * D; }
  else if (mat < 2 * nrel)   { r = rel + (size_t)(mat - nrel) * D; W = out_w;  out = Mmat + (size_t)mat * D * D; }
  else                       { r = loop_rel;                       W = loop_w; out = Mloop; }
  __shared__ float rl[D];
  rl[threadIdx.x] = r[threadIdx.x];
  __syncthreads();
  int c = threadIdx.x;        // N index
  float acc = 0.f;
  for (int k = 0; k < D; ++k)
    acc += rl[(j + k) & 127] * W[k * D + c];
  out[bfragIdx(j, c)] = acc;  // fragment-order store
}

// ------------------------------------- main edge kernel: WMMA + atomic scatter
__global__ __launch_bounds__(32) void edge_wmma_kernel(
    const float* __restrict__ x, const float* __restrict__ Mmat,
    const int* __restrict__ sorted, const Tile* __restrict__ tiles,
    const int* __restrict__ numTiles,
    const int* __restrict__ edge_src, const int* __restrict__ edge_dst,
    const float* __restrict__ edge_norm, float* __restrict__ agg) {
  int tid = blockIdx.x;
  if (tid >= *numTiles) return;
  Tile t = tiles[tid];

  __shared__ float Alds[32 * D];
  __shared__ float normL[32];
  __shared__ int dstL[32];
  __shared__ int srcL[32];

  int lane = threadIdx.x;
  int e = (lane < t.cnt) ? sorted[t.start + lane] : -1;
  srcL[lane]  = (e >= 0) ? edge_src[e] : -1;
  dstL[lane]  = (e >= 0) ? edge_dst[e] : 0;
  normL[lane] = (e >= 0) ? edge_norm[e] : 0.f;
  __syncthreads();

  // stage gathered A tile (32 edges x 128 feats) via coalesced float4 loads
  for (int row = 0; row < 32; ++row) {
    int s = srcL[row];
    float4 v = make_float4(0.f, 0.f, 0.f, 0.f);
    if (s >= 0) v = ((const float4*)(x + (size_t)s * D))[lane];
    ((float4*)(Alds + row * D))[lane] = v;
  }
  __syncthreads();

  const v2f* __restrict__ Bf = (const v2f*)(Mmat + (size_t)t.key * D * D);
  const v2f* __restrict__ A2 = (const v2f*)Alds;
  int hi = lane >> 4;
  int ln = lane & 15;

  for (int nb = 0; nb < 8; ++nb) {
    v8f acc0 = {};
    v8f acc1 = {};
#pragma unroll
    for (int kb = 0; kb < 32; ++kb) {
      // one coalesced 8B load per lane: whole wave reads 256B contiguous
      v2f b  = Bf[(kb * 8 + nb) * 32 + lane];
      // A: lanes0-15 K={4kb,4kb+1}, lanes16-31 K={4kb+2,4kb+3} -> ds_load_b64
      v2f a0 = A2[ln * 64 + kb * 2 + hi];
      v2f a1 = A2[(16 + ln) * 64 + kb * 2 + hi];
      acc0 = __builtin_amdgcn_wmma_f32_16x16x4_f32(false, a0, false, b,
                                                   (short)0, acc0, false, false);
      acc1 = __builtin_amdgcn_wmma_f32_16x16x4_f32(false, a1, false, b,
                                                   (short)0, acc1, false, false);
    }
    int col = nb * 16 + ln;
#pragma unroll
    for (int v = 0; v < 8; ++v) {
      int m0 = v + 8 * hi;       // D layout: VGPR v -> M=v (lanes0-15) / v+8
      if (m0 < t.cnt)
        unsafeAtomicAdd(&agg[(size_t)dstL[m0] * D + col], acc0[v] * normL[m0]);
      int m1 = m0 + 16;
      if (m1 < t.cnt)
        unsafeAtomicAdd(&agg[(size_t)dstL[m1] * D + col], acc1[v] * normL[m1]);
    }
  }
}

// ------------------------------------------------- self-loop: x @ Mloop
__global__ __launch_bounds__(32) void loop_wmma_kernel(
    const float* __restrict__ x, const float* __restrict__ Mloop,
    float* __restrict__ agg, int V) {
  int base = blockIdx.x * 32;
  __shared__ float Alds[32 * D];
  int lane = threadIdx.x;
  for (int row = 0; row < 32; ++row) {
    int r = base + row;
    float4 v = make_float4(0.f, 0.f, 0.f, 0.f);
    if (r < V) v = ((const float4*)(x + (size_t)r * D))[lane];
    ((float4*)(Alds + row * D))[lane] = v;
  }
  __syncthreads();
  const v2f* __restrict__ Bf = (const v2f*)Mloop;
  const v2f* __restrict__ A2 = (const v2f*)Alds;
  int hi = lane >> 4;
  int ln = lane & 15;
  for (int nb = 0; nb < 8; ++nb) {
    v8f acc0 = {};
    v8f acc1 = {};
#pragma unroll
    for (int kb = 0; kb < 32; ++kb) {
      v2f b  = Bf[(kb * 8 + nb) * 32 + lane];
      v2f a0 = A2[ln * 64 + kb * 2 + hi];
      v2f a1 = A2[(16 + ln) * 64 + kb * 2 + hi];
      acc0 = __builtin_amdgcn_wmma_f32_16x16x4_f32(false, a0, false, b,
                                                   (short)0, acc0, false, false);
      acc1 = __builtin_amdgcn_wmma_f32_16x16x4_f32(false, a1, false, b,
                                                   (short)0, acc1, false, false);
    }
    int col = nb * 16 + ln;
#pragma unroll
    for (int v = 0; v < 8; ++v) {
      int r0 = base + v + 8 * hi;
      if (r0 < V) agg[(size_t)r0 * D + col] += acc0[v];
      int r1 = r0 + 16;
      if (r1 < V) agg[(size_t)r1 * D + col] += acc1[v];
    }
  }
}

// ------------------------------------------------------------- BatchNorm
__global__ __launch_bounds__(128) void colsum_kernel(const float* __restrict__ h,
                                                     int V, float* __restrict__ S,
                                                     float* __restrict__ Q) {
  int c = threadIdx.x;
  float s = 0.f, q = 0.f;
  for (int r = blockIdx.x; r < V; r += gridDim.x) {
    float a = h[(size_t)r * D + c];
    s += a;
    q += a * a;
  }
  unsafeAtomicAdd(&S[c], s);
  unsafeAtomicAdd(&Q[c], q);
}

// h = a/3 + bias; BN over nodes. bias cancels; /3 folds into scale.
__global__ void bnfin_kernel(const float* __restrict__ S, const float* __restrict__ Q,
                             const float* __restrict__ gamma, const float* __restrict__ beta,
                             float Vf, float* __restrict__ scale, float* __restrict__ shift) {
  int c = threadIdx.x;
  float mean = S[c] / Vf;
  float var_a = Q[c] / Vf - mean * mean;             // var of agg
  float inv = rsqrtf(var_a * (1.0f / 9.0f) + 1e-5f); // var_h = var_a/9
  float sc = gamma[c] * inv * (1.0f / 3.0f);
  scale[c] = sc;
  shift[c] = beta[c] - mean * sc;
}

__global__ void bnapply_kernel(float* __restrict__ h, long n,
                               const float* __restrict__ scale,
                               const float* __restrict__ shift) {
  long i = (long)blockIdx.x * blockDim.x + threadIdx.x;
  long st = (long)gridDim.x * blockDim.x;
  for (; i < n; i += st) {
    int c = (int)(i & (D - 1));
    h[i] = h[i] * scale[c] + shift[c];
  }
}

// ------------------------------------------------------------- rel @ w_rel
__global__ __launch_bounds__(128) void relproj_kernel(const float* __restrict__ rel,
                                                      const float* __restrict__ w_rel,
                                                      float* __restrict__ out2) {
  int t = blockIdx.x;
  int c = threadIdx.x;
  __shared__ float rl[D];
  rl[c] = rel[(size_t)t * D + c];
  __syncthreads();
  float acc = 0.f;
  for (int k = 0; k < D; ++k) acc += rl[k] * w_rel[k * D + c];
  out2[(size_t)t * D + c] = acc;
}

// ----------------------------------------------------------------- launcher
extern "C" void kernel_launch(void* const* d_in, const int* in_sizes, int n_in,
                              void* d_out, int out_size, void* d_ws, size_t ws_size,
                              hipStream_t stream) {
  const float* x         = (const float*)d_in[0];
  const float* rel       = (const float*)d_in[1];
  const float* edge_norm = (const float*)d_in[2];
  const float* in_w      = (const float*)d_in[3];
  const float* out_w     = (const float*)d_in[4];
  const float* loop_w    = (const float*)d_in[5];
  const float* w_rel     = (const float*)d_in[6];
  const float* loop_rel  = (const float*)d_in[7];
  // d_in[8] bias: zeros + cancels under BN (folded)
  const float* bn_gamma  = (const float*)d_in[9];
  const float* bn_beta   = (const float*)d_in[10];
  const int* edge_src    = (const int*)d_in[11];
  const int* edge_dst    = (const int*)d_in[12];
  const int* edge_type   = (const int*)d_in[13];

  const int V    = in_sizes[0] / D;
  const int nrel = in_sizes[1] / D;   // 100 (NREL2)
  const int E    = in_sizes[2];
  const int half = E / 2;
  const int nkey = 2 * nrel;          // 200

  // ---- workspace layout
  char* wsb = (char*)d_ws;
  float* S      = (float*)wsb;        // [128]
  float* Q      = S + D;              // [128]
  float* scaleA = Q + D;              // [128]
  float* shiftA = scaleA + D;         // [128]
  int* cnt      = (int*)(wsb + 4096); // [256]
  int* cursor   = cnt + 256;
  int* offsets  = cursor + 256;
  int* tbase    = offsets + 256;
  int* numTiles = tbase + 256;
  Tile* tiles   = (Tile*)(wsb + 16384);
  const int maxTiles = E / 32 + nkey + 1;
  size_t off = 16384 + (((size_t)maxTiles * sizeof(Tile) + 255) & ~(size_t)255);
  int* sorted = (int*)(wsb + off);
  off += (((size_t)E * 4 + 255) & ~(size_t)255);
  float* Mmat = (float*)(wsb + off);
  float* Mloop = Mmat + (size_t)nkey * D * D;

  float* agg = (float*)d_out;                 // h accumulates here, BN in place
  float* out2 = (float*)d_out + (size_t)V * D;

  // 1) zero accumulators + metadata
  zero_f32<<<512, 256, 0, stream>>>(agg, (long)V * D);
  zero_f32<<<16, 256, 0, stream>>>((float*)wsb, 4096);

  // 2) bucket-sort edges by (dir,type)
  hist_kernel<<<(E + 255) / 256, 256, 0, stream>>>(edge_type, E, half, nrel, cnt);
  scan_kernel<<<1, 32, 0, stream>>>(cnt, nkey, offsets, tbase, numTiles);
  tiles_kernel<<<nkey, 128, 0, stream>>>(cnt, offsets, tbase, tiles);
  scatter_kernel<<<(E + 255) / 256, 256, 0, stream>>>(edge_type, E, half, nrel,
                                                      offsets, cursor, sorted);

  // 3) fused relation matrices M = C(rel) @ W (+ Mloop), B-fragment order
  buildM_kernel<<<(nkey + 1) * D, 128, 0, stream>>>(rel, in_w, out_w, loop_rel,
                                                    loop_w, nrel, Mmat, Mloop);

  // 4) per-edge matvec via fp32 WMMA + atomic scatter-add
  edge_wmma_kernel<<<maxTiles, 32, 0, stream>>>(x, Mmat, sorted, tiles, numTiles,
                                                edge_src, edge_dst, edge_norm, agg);

  // 5) self-loop term
  loop_wmma_kernel<<<(V + 31) / 32, 32, 0, stream>>>(x, Mloop, agg, V);

  // 6) BatchNorm (two-pass, in place)
  colsum_kernel<<<512, 128, 0, stream>>>(agg, V, S, Q);
  bnfin_kernel<<<1, 128, 0, stream>>>(S, Q, bn_gamma, bn_beta, (float)V,
                                      scaleA, shiftA);
  bnapply_kernel<<<2048, 256, 0, stream>>>(agg, (long)V * D, scaleA, shiftA);

  // 7) relation projection
  relproj_kernel<<<nrel, 128, 0, stream>>>(rel, w_rel, out2);
}